// Maploss_63110249447665
// MI455X (gfx1250) — compile-verified
//
#include <hip/hip_runtime.h>
#include <stdint.h>

// ---------------------------------------------------------------------------
// OHEM loss for MI455X (gfx1250).
// 268 MB streamed once -> ~11.5 us at 23.3 TB/s: pure bandwidth problem.
// - TDM (tensor_load_to_lds) stages 8KB label/pred tiles into LDS,
//   DOUBLE-BUFFERED: chunk c+1 is in flight while chunk c is processed;
//   s_wait_tensorcnt(2) retires the oldest pair (in-order per wave).
// - Top-k-of-negatives solved with a 512-bin float-order histogram
//   (monotone bit pattern of non-negative f32), exact when k >= N.
// - Wave reduction of (pos_sum, pos_count) done on the matrix pipe via
//   V_WMMA_F32_16X16X4_F32 with a column-selector B matrix (full f32).
// ---------------------------------------------------------------------------

#define HW        (512 * 512)
#define BATCH     32
#define NROWS     128                 // 4 components * 32 samples
#define NBINS     512
#define CHUNK     2048                // floats per TDM tile (8 KB)
#define BLOCKS_PER_ROW 16
#define SPAN      (HW / BLOCKS_PER_ROW)   // 16384
#define NCHUNKS   (SPAN / CHUNK)          // 8
#define THREADS   256
#define THRESH_F  0.1f
#define TOPK_NO_POS 500

typedef __attribute__((ext_vector_type(2))) float        v2f;
typedef __attribute__((ext_vector_type(8))) float        v8f;
typedef __attribute__((ext_vector_type(4))) unsigned int u32x4;
typedef __attribute__((ext_vector_type(4))) int          i32x4;
typedef __attribute__((ext_vector_type(8))) int          i32x8;

#if __has_builtin(__builtin_amdgcn_tensor_load_to_lds)
#define HAVE_TDM 1
#else
#define HAVE_TDM 0
#endif

// Workspace layout (dwords):
//   [0, NROWS*NBINS)                     : per-row per-bin negative-loss sums (f32)
//   [NROWS*NBINS, 2*NROWS*NBINS)         : per-row per-bin negative counts   (u32)
//   [2*NROWS*NBINS, +NROWS)              : per-row positive-loss sum         (f32)
//   [2*NROWS*NBINS+NROWS, +NROWS)        : per-row positive count            (u32)
#define WS_DWORDS (2 * NROWS * NBINS + 2 * NROWS)

__global__ __launch_bounds__(256) void ohem_zero(unsigned* __restrict__ ws,
                                                 float* __restrict__ out) {
    int i = blockIdx.x * blockDim.x + threadIdx.x;
    const int stride = gridDim.x * blockDim.x;
    for (; i < WS_DWORDS; i += stride) ws[i] = 0u;
    if (blockIdx.x == 0 && threadIdx.x == 0) out[0] = 0.0f;
}

#if HAVE_TDM
// Issue one TDM load of CHUNK contiguous f32 from global into LDS.
// D# built per CDNA5 ISA sec. 8 (1-row tile, data_size=4B, no pad/iterate).
__device__ __forceinline__ void tdm_tile(unsigned lds_byte, unsigned long long gaddr) {
    u32x4 g0 = {
        1u,                                                    // count=1, user desc
        lds_byte,                                              // lds_addr [63:32]
        (unsigned)(gaddr & 0xffffffffull),                     // global_addr lo
        (unsigned)((gaddr >> 32) & 0x01ffffffull) | (2u << 30) // addr hi | type=2
    };
    i32x8 g1 = {
        (int)(2u << 16),                          // data_size = 4 bytes
        (int)(((unsigned)CHUNK & 0xffffu) << 16), // tensor_dim0 lo16 @ [31:16]
        (int)(((unsigned)CHUNK >> 16) | (1u << 16)), // tensor_dim0 hi | tensor_dim1=1
        (int)(((unsigned)CHUNK & 0xffffu) << 16), // tile_dim0 @ [31:16]
        1,                                        // tile_dim1 = 1, tile_dim2 = 0
        (int)CHUNK,                               // tensor_dim0_stride lo32
        0, 0
    };
    i32x4 gz = {0, 0, 0, 0};
#if __has_include(<hip/amd_detail/amd_gfx1250_TDM.h>)
    i32x8 gz8 = {0, 0, 0, 0, 0, 0, 0, 0};
    __builtin_amdgcn_tensor_load_to_lds(g0, g1, gz, gz, gz8, 0);
#else
    __builtin_amdgcn_tensor_load_to_lds(g0, g1, gz, gz, 0);
#endif
}
#endif

__global__ __launch_bounds__(THREADS) void ohem_stats(
    const float* __restrict__ l0, const float* __restrict__ l1,
    const float* __restrict__ l2, const float* __restrict__ l3,
    const float* __restrict__ p0, const float* __restrict__ p1,
    const float* __restrict__ p2, const float* __restrict__ p3,
    float* __restrict__ ws) {
    __shared__ float    lds_l[2][CHUNK];
    __shared__ float    lds_p[2][CHUNK];
    __shared__ unsigned h_cnt[NBINS];
    __shared__ float    h_sum[NBINS];

    const int row  = blockIdx.x;       // 0..127
    const int comp = row >> 5;
    const int samp = row & 31;
    const float* lab = (comp == 0) ? l0 : (comp == 1) ? l1 : (comp == 2) ? l2 : l3;
    const float* prd = (comp == 0) ? p0 : (comp == 1) ? p1 : (comp == 2) ? p2 : p3;
    const size_t base = (size_t)samp * HW + (size_t)blockIdx.y * SPAN;

#if HAVE_TDM
    if (threadIdx.x < 32) {  // prologue: chunk 0 into buffer 0
        tdm_tile((unsigned)(unsigned long long)(size_t)&lds_l[0][0],
                 (unsigned long long)(size_t)(lab + base));
        tdm_tile((unsigned)(unsigned long long)(size_t)&lds_p[0][0],
                 (unsigned long long)(size_t)(prd + base));
    }
#endif

    for (int i = threadIdx.x; i < NBINS; i += THREADS) { h_cnt[i] = 0u; h_sum[i] = 0.0f; }

    float    p_sum = 0.0f;
    unsigned p_cnt = 0u;

    for (int c = 0; c < NCHUNKS; ++c) {
        const int buf = c & 1;
#if HAVE_TDM
        if (threadIdx.x < 32) {  // wave 0 drives the Tensor Data Mover
            if (c + 1 < NCHUNKS) {
                const size_t noff = base + (size_t)(c + 1) * CHUNK;
                const int    nbuf = (c + 1) & 1;
                tdm_tile((unsigned)(unsigned long long)(size_t)&lds_l[nbuf][0],
                         (unsigned long long)(size_t)(lab + noff));
                tdm_tile((unsigned)(unsigned long long)(size_t)&lds_p[nbuf][0],
                         (unsigned long long)(size_t)(prd + noff));
                // 4 outstanding; oldest pair (chunk c) retires in-order
                __builtin_amdgcn_s_wait_tensorcnt(2);
            } else {
                __builtin_amdgcn_s_wait_tensorcnt(0);
            }
        }
#else
        {
            const size_t off = base + (size_t)c * CHUNK;
            for (int i = threadIdx.x; i < CHUNK / 4; i += THREADS) {
                ((float4*)lds_l[buf])[i] = ((const float4*)(lab + off))[i];
                ((float4*)lds_p[buf])[i] = ((const float4*)(prd + off))[i];
            }
        }
#endif
        __syncthreads();
#pragma unroll
        for (int i = 0; i < CHUNK / (4 * THREADS); ++i) {  // 2x ds_load_b128 each
            const int    t  = threadIdx.x + i * THREADS;
            const float4 lv = ((const float4*)lds_l[buf])[t];
            const float4 pv = ((const float4*)lds_p[buf])[t];
#pragma unroll
            for (int e = 0; e < 4; ++e) {
                const float lvx  = (&lv.x)[e];
                const float d    = (&pv.x)[e] - lvx;
                const float loss = d * d;
                if (lvx >= THRESH_F) {
                    p_cnt++;
                    p_sum += loss;
                } else {
                    // non-negative f32: bit pattern is order-preserving
                    const unsigned bin = (__float_as_uint(loss) >> 22) & (NBINS - 1);
                    atomicAdd(&h_cnt[bin], 1u);
                    atomicAdd(&h_sum[bin], loss);
                }
            }
        }
        __syncthreads();
    }

    float*    ws_sum    = ws;
    unsigned* ws_cnt    = (unsigned*)(ws + NROWS * NBINS);
    float*    ws_possum = ws + 2 * NROWS * NBINS;
    unsigned* ws_P      = (unsigned*)(ws + 2 * NROWS * NBINS + NROWS);

    // ---- wave reduction of (p_sum, p_cnt) on the matrix pipe --------------
    // A[m][k]: lane L holds (p_sum, p_cnt) as K-pair; B column 0 selects even
    // K (sum), column 1 selects odd K (count). EXEC is all-ones here.
#if __has_builtin(__builtin_amdgcn_wmma_f32_16x16x4_f32)
    {
        const int lane = threadIdx.x & 31;
        v2f a;  a.x = p_sum;  a.y = (float)p_cnt;
        v2f bm; bm.x = ((lane & 15) == 0) ? 1.0f : 0.0f;
                bm.y = ((lane & 15) == 1) ? 1.0f : 0.0f;
        v8f acc = {};
        acc = __builtin_amdgcn_wmma_f32_16x16x4_f32(false, a, false, bm,
                                                    (short)0, acc, false, false);
        const float tot = acc[0] + acc[1] + acc[2] + acc[3] +
                          acc[4] + acc[5] + acc[6] + acc[7];
        if ((lane & 15) == 0) {            // lanes 0,16: the two p_sum halves
            atomicAdd(&ws_possum[row], tot);
        } else if ((lane & 15) == 1) {     // lanes 1,17: the two count halves
            atomicAdd(&ws_P[row], (unsigned)(tot + 0.5f));
        }
    }
#else
    {
        float s = p_sum, cf = (float)p_cnt;
        for (int o = 16; o; o >>= 1) {
            s  += __shfl_down(s, o, 32);
            cf += __shfl_down(cf, o, 32);
        }
        if ((threadIdx.x & 31) == 0) {
            atomicAdd(&ws_possum[row], s);
            atomicAdd(&ws_P[row], (unsigned)(cf + 0.5f));
        }
    }
#endif

    // ---- flush block-local histogram into the per-row global one ----------
    for (int i = threadIdx.x; i < NBINS; i += THREADS) {
        const unsigned cv = h_cnt[i];
        if (cv) {
            atomicAdd(&ws_cnt[row * NBINS + i], cv);
            atomicAdd(&ws_sum[row * NBINS + i], h_sum[i]);
        }
    }
}

__global__ __launch_bounds__(NROWS) void ohem_finalize(const float* __restrict__ ws,
                                                       float* __restrict__ out) {
    const int row = threadIdx.x;
    if (row >= NROWS) return;
    const float*    bsum = ws + (size_t)row * NBINS;
    const unsigned* bcnt = (const unsigned*)(ws + NROWS * NBINS) + (size_t)row * NBINS;
    const float     pos_sum = ws[2 * NROWS * NBINS + row];
    const unsigned  P = ((const unsigned*)(ws + 2 * NROWS * NBINS + NROWS))[row];

    unsigned N = 0;
    float    negtot = 0.0f;
    for (int i = 0; i < NBINS; ++i) { N += bcnt[i]; negtot += bsum[i]; }

    unsigned k = (P > 0) ? min(3u * P, N) : (unsigned)min(TOPK_NO_POS, HW);

    float topk = negtot;                    // exact when k >= N (common case)
    if (k < N) {                            // suffix scan over descending bins
        unsigned cum = 0;
        float    shi = 0.0f;
        topk = 0.0f;
        for (int b = NBINS - 1; b >= 0; --b) {
            const unsigned c = bcnt[b];
            if (cum + c >= k) {
                const unsigned need = k - cum;
                topk = shi + (c ? bsum[b] * ((float)need / (float)c) : 0.0f);
                break;
            }
            cum += c;
            shi += bsum[b];
        }
    }

    float per;
    if (P > 0) {
        const float posi = pos_sum / (float)P;
        const float nega = (N > 0) ? (topk / (float)(k ? k : 1u)) : 0.0f;
        per = posi + nega;
    } else {
        per = topk / (float)k;              // top-500 branch, k == 500
    }
    atomicAdd(out, per * (1.0f / (float)BATCH));
}

extern "C" void kernel_launch(void* const* d_in, const int* in_sizes, int n_in,
                              void* d_out, int out_size, void* d_ws, size_t ws_size,
                              hipStream_t stream) {
    (void)in_sizes; (void)n_in; (void)out_size; (void)ws_size;
    const float* l0 = (const float*)d_in[0];  // gh_label
    const float* l1 = (const float*)d_in[1];  // gah_label
    const float* l2 = (const float*)d_in[2];  // ori_x
    const float* l3 = (const float*)d_in[3];  // ori_y
    const float* p0 = (const float*)d_in[4];  // p_gh
    const float* p1 = (const float*)d_in[5];  // p_gah
    const float* p2 = (const float*)d_in[6];  // p_ori_x
    const float* p3 = (const float*)d_in[7];  // p_ori_y
    float* ws  = (float*)d_ws;
    float* out = (float*)d_out;

    ohem_zero<<<256, 256, 0, stream>>>((unsigned*)ws, out);
    dim3 grid(NROWS, BLOCKS_PER_ROW);
    ohem_stats<<<grid, THREADS, 0, stream>>>(l0, l1, l2, l3, p0, p1, p2, p3, ws);
    ohem_finalize<<<1, NROWS, 0, stream>>>(ws, out);
}